// Superfeatures_77129022701763
// MI455X (gfx1250) — compile-verified
//
#include <hip/hip_runtime.h>

#define NSEG   512
#define NCH    64
#define HW     512
#define IH     510
#define NPIX   (IH * IH)            // 260100 interior pixels
#define TPB    256
#define PPT    16                   // pixels per thread
#define CHUNK  (TPB * PPT)          // 4096 pixels per block
#define NCHUNK 64                   // 64*4096 = 262144 >= 260100

#define ACC_ELEMS (NCH * NSEG)      // 32768 floats = 128 KB
#define ACC_BYTES (ACC_ELEMS * 4)
#define IDX_BYTES (CHUNK * 4)       // int32 staging = 16 KB
#define LDS_BYTES (ACC_BYTES + IDX_BYTES)   // 144 KB (CDNA5 WGP has 320 KB)

__global__ __launch_bounds__(TPB) void segmax_init(float* __restrict__ out, int n) {
    int i = blockIdx.x * TPB + threadIdx.x;
    if (i < n) out[i] = -__builtin_inff();
}

__global__ __launch_bounds__(TPB) void segmax_acc(const float* __restrict__ feats,
                                                  const int* __restrict__ labels,
                                                  float* __restrict__ out) {
    extern __shared__ char lds[];
    float* acc = (float*)lds;

    const int tid  = (int)threadIdx.x;
    const int b    = (int)blockIdx.y;
    const int base = (int)blockIdx.x * CHUNK;

    // Byte offsets into LDS for inline-asm DS/async ops (dynamic LDS starts
    // after static LDS; none here, but be exact):
    const unsigned ldsBase = __builtin_amdgcn_groupstaticsize();
    const unsigned accOff  = ldsBase;
    const unsigned idxOff  = ldsBase + ACC_BYTES;

    // ---- init private accumulator to -inf (normal DS stores, compiler-tracked)
    for (int i = tid; i < ACC_ELEMS; i += TPB) acc[i] = -__builtin_inff();

    // ---- async-stage the int32 label chunk into LDS (reused 64x, once per channel)
    const int* labB = labels + (long long)b * (HW * HW);
    int goff[PPT];
    int valid[PPT];
#pragma unroll
    for (int k = 0; k < PPT; ++k) {
        int pix = base + k * TPB + tid;
        int v   = pix < NPIX;
        int p   = v ? pix : (NPIX - 1);
        int h   = p / IH;
        int w   = p - h * IH;
        int off = (h + 1) * HW + (w + 1);   // interior -> full-image offset
        goff[k]  = off;
        valid[k] = v;
        unsigned laddr = idxOff + (unsigned)(k * TPB + tid) * 4u;
        const int* gaddr = labB + off;
        asm volatile("global_load_async_to_lds_b32 %0, %1, off"
                     :: "v"(laddr), "v"(gaddr) : "memory");
    }
    asm volatile("s_wait_asynccnt 0" ::: "memory");
    __syncthreads();   // also covers accumulator init

    // ---- pull this thread's 16 segment ids into regs
    int seg[PPT];
    const int* idx32 = (const int*)(lds + ACC_BYTES);
#pragma unroll
    for (int k = 0; k < PPT; ++k) seg[k] = idx32[k * TPB + tid];

    // ---- stream features channel by channel, LDS-atomic-max into acc[c][s]
    // layout [c*512 + s]: bank = s mod 64 -> random segments hit all 64 banks
    const float* fB = feats + ((long long)b * NCH) * (HW * HW);
    for (int c = 0; c < NCH; ++c) {
        const float* fc = fB + (long long)c * (HW * HW);
        const unsigned accC = accOff + (unsigned)(c * NSEG) * 4u;
#pragma unroll
        for (int k = 0; k < PPT; ++k) {
            if (valid[k]) {
                float v = fc[goff[k]];                       // coalesced b32 stream
                unsigned a = accC + (unsigned)seg[k] * 4u;   // LDS byte address
                asm volatile("ds_max_num_f32 %0, %1" :: "v"(a), "v"(v) : "memory");
            }
        }
    }

    // our asm DS atomics are invisible to the compiler's waitcnt insertion
    asm volatile("s_wait_dscnt 0" ::: "memory");
    __syncthreads();

    // ---- flush private accumulator to global with hardware f32 atomic max
    float* outB = out + (long long)b * ACC_ELEMS;
    for (int i = tid; i < ACC_ELEMS; i += TPB) {
        float v = acc[i];
        if (v > -__builtin_inff()) {        // skip untouched segments
            float* addr = outB + i;
            asm volatile("global_atomic_max_num_f32 %0, %1, off"
                         :: "v"(addr), "v"(v) : "memory");
        }
    }
}

__global__ __launch_bounds__(TPB) void segmax_fin(float* __restrict__ out, int n) {
    int i = blockIdx.x * TPB + threadIdx.x;
    if (i < n) {
        float v = out[i];
        out[i] = __builtin_isfinite(v) ? v : 0.0f;   // empty segments (-inf) -> 0
    }
}

extern "C" void kernel_launch(void* const* d_in, const int* in_sizes, int n_in,
                              void* d_out, int out_size, void* d_ws, size_t ws_size,
                              hipStream_t stream) {
    const float* feats  = (const float*)d_in[0];   // (8, 64, 512, 512) f32
    const int*   labels = (const int*)d_in[1];     // (8, 1, 512, 512) int32 per harness convention
    float* out = (float*)d_out;                    // (8, 64, 512) f32

    const int n = 8 * NCH * NSEG;                  // 262144
    segmax_init<<<dim3((n + TPB - 1) / TPB), TPB, 0, stream>>>(out, n);
    segmax_acc <<<dim3(NCHUNK, 8), TPB, LDS_BYTES, stream>>>(feats, labels, out);
    segmax_fin <<<dim3((n + TPB - 1) / TPB), TPB, 0, stream>>>(out, n);
}